// ColorMatchingLoss_46531675685364
// MI455X (gfx1250) — compile-verified
//
#include <hip/hip_runtime.h>
#include <hip/hip_bf16.h>

typedef __attribute__((ext_vector_type(2))) float v2f;
typedef __attribute__((ext_vector_type(8))) float v8f;

#define NSAMP   6144
#define ROWS    12288      // 2 * NSAMP
#define KDIM    64         // batch
#define NT      768        // 12288 / 16 tiles per dimension
#define NSUB    4          // N-subtiles per wave (16x64 strip)
#define HDIM    256
#define WDIM    256
#define FLT_BIG 3.402823466e38f

// ---- order-preserving float <-> uint key (so atomicMin(u32) == float min) ----
__device__ __forceinline__ unsigned fkey(float f) {
    unsigned u = __float_as_uint(f);
    return (u & 0x80000000u) ? ~u : (u | 0x80000000u);
}
__device__ __forceinline__ float fkey_inv(unsigned k) {
    unsigned u = (k & 0x80000000u) ? (k & 0x7FFFFFFFu) : ~k;
    return __uint_as_float(u);
}

// ---- stage 0: init min buffers to +inf key ----
__global__ void init_min_kernel(unsigned* __restrict__ mins, int n) {
    int i = blockIdx.x * blockDim.x + threadIdx.x;
    if (i < n) mins[i] = 0xFFFFFFFFu;
}

// ---- stage 1: gather sampled pixels, RGB->YUV (U or V row), write M (ROWS x 64),
// per-row sum of squares, and per-row 1/(sqrt(sq)+eps). One wave per row.
__global__ __launch_bounds__(256) void gather_yuv_kernel(
    const float* __restrict__ img,          // (64,3,256,256)
    const int*   __restrict__ iy,
    const int*   __restrict__ ix,
    float* __restrict__ M,                  // (ROWS, 64)
    float* __restrict__ sq,                 // (ROWS)
    float* __restrict__ inv)                // (ROWS): 1/(sqrt(sq)+1e-16)
{
    const int lane = threadIdx.x & 31;
    const int wave = threadIdx.x >> 5;
    const int i    = blockIdx.x * 8 + wave;     // row index, exact fit (1536*8)
    const int c    = i / NSAMP;                 // 0 -> U row, 1 -> V row
    const int n    = i - c * NSAMP;

    const int y = iy[n];
    const int x = ix[n];
    const size_t pix = (size_t)y * WDIM + x;

    const float w0 = (c == 0) ? -0.14714119f : 0.61497538f;
    const float w1 = (c == 0) ? -0.28886916f : -0.51496512f;
    const float w2 = (c == 0) ?  0.43601035f : -0.10001026f;

    float s = 0.0f;
#pragma unroll
    for (int t = 0; t < 2; ++t) {
        const int b = lane + 32 * t;
        const size_t base = (size_t)b * 3 * (HDIM * WDIM) + pix;
        const float r  = (img[base]                   + 1.0f) * 0.5f;
        const float g  = (img[base + HDIM * WDIM]     + 1.0f) * 0.5f;
        const float bl = (img[base + 2 * HDIM * WDIM] + 1.0f) * 0.5f;
        const float v  = w0 * r + w1 * g + w2 * bl;
        M[(size_t)i * KDIM + b] = v;
        s += v * v;
    }
#pragma unroll
    for (int off = 1; off < 32; off <<= 1) s += __shfl_xor(s, off, 32);
    if (lane == 0) {
        sq[i]  = s;
        inv[i] = 1.0f / (sqrtf(s) + 1e-16f);
    }
}

// ---- stage 2: fused WMMA GEMM (f32 16x16x4) + distance + row/col min atomics.
// One wave computes a 16x64 strip (4 subtiles of 16x16), K=64.
// A fragments + x-norm terms loaded once and reused across the 4 subtiles.
// Epilogue is divide/sqrt-free: all reciprocals precomputed per row.
__global__ __launch_bounds__(256) void gemm_dist_kernel(
    const float* __restrict__ X,            // (ROWS, 64)
    const float* __restrict__ Y,            // (ROWS, 64)
    const float* __restrict__ xsq,          // (ROWS)
    const float* __restrict__ ysq,          // (ROWS)
    const float* __restrict__ xinv,         // (ROWS)
    const float* __restrict__ yinv,         // (ROWS)
    unsigned* __restrict__ rowmin,          // (ROWS) keys
    unsigned* __restrict__ colmin)          // (ROWS) keys
{
    const int lane   = threadIdx.x & 31;
    const int waveIb = threadIdx.x >> 5;
    const long wId   = (long)blockIdx.x * 8 + waveIb;   // NT * (NT/NSUB) waves
    const int mt = (int)(wId / (NT / NSUB));
    const int g  = (int)(wId % (NT / NSUB));
    const int m0 = mt * 16;

    const int hi  = lane >> 4;      // half-wave select
    const int l15 = lane & 15;

    // A (16x4 f32): lane L<16 holds X[m0+L][k..k+1]; L>=16 holds X[m0+L-16][k+2..k+3]
    const float* xrow = X + (size_t)(m0 + l15) * KDIM + 2 * hi;

    // Preload all 16 A fragments into distinct registers (loads pipeline as a clause).
    v2f a[16];
#pragma unroll
    for (int k = 0; k < 16; ++k) a[k] = *(const v2f*)(xrow + 4 * k);

    // Per-row x terms, loaded once per wave.
    float xs[8], xin[8];
#pragma unroll
    for (int r = 0; r < 8; ++r) {
        xs[r]  = xsq[m0 + r + 8 * hi];
        xin[r] = xinv[m0 + r + 8 * hi];
    }

    // Register-resident row mins across subtiles.
    float rowPart[8];
#pragma unroll
    for (int r = 0; r < 8; ++r) rowPart[r] = FLT_BIG;

    for (int s = 0; s < NSUB; ++s) {
        const int n0 = (g * NSUB + s) * 16;
        const float* yrow = Y + (size_t)(n0 + l15) * KDIM + 2 * hi;

        // B (4x16 f32): lane L<16 holds Y[n0+L][k..k+1]; L>=16 holds Y[n0+L-16][k+2..k+3]
        v2f b[16];
#pragma unroll
        for (int k = 0; k < 16; ++k) b[k] = *(const v2f*)(yrow + 4 * k);

        // Prefetch next subtile's B rows (global_prefetch_b8, near-cache hint).
        if (s + 1 < NSUB) {
            const float* ynext = Y + (size_t)(n0 + 16 + l15) * KDIM + 2 * hi;
            __builtin_prefetch(ynext, 0, 3);
        }

        v8f acc = {0.f, 0.f, 0.f, 0.f, 0.f, 0.f, 0.f, 0.f};
#pragma unroll
        for (int k = 0; k < 16; ++k) {
            acc = __builtin_amdgcn_wmma_f32_16x16x4_f32(
                /*neg_a=*/false, a[k], /*neg_b=*/false, b[k],
                /*c_mod=*/(short)0, acc, /*reuse_a=*/false, /*reuse_b=*/false);
        }

        // C/D layout: VGPR r, lanes 0-15 -> (M=r, N=lane); lanes 16-31 -> (M=8+r, N=lane-16)
        const float ysq_n = ysq[n0 + l15];
        const float yin_n = yinv[n0 + l15];

        float colPart = FLT_BIG;
#pragma unroll
        for (int r = 0; r < 8; ++r) {
            const float xy = acc[r];
            float l2 = xs[r] + ysq_n - 2.0f * xy;
            l2 = fminf(fmaxf(l2, 1e-5f), 1e5f) * (1.0f / (float)KDIM);
            const float d = l2 + 1.0f - xy * xin[r] * yin_n;
            rowPart[r] = fminf(rowPart[r], d);
            colPart    = fminf(colPart, d);
        }

        // Col mins: combine the two halves (M 0-7 with M 8-15), lanes 0-15 commit.
        const float c2 = fminf(colPart, __shfl_xor(colPart, 16, 32));
        if (hi == 0) atomicMin(&colmin[n0 + l15], fkey(c2));
    }

    // Row mins: reduce over N (16 lanes within each half), lanes 0 & 16 commit once.
#pragma unroll
    for (int r = 0; r < 8; ++r) {
        float v = rowPart[r];
#pragma unroll
        for (int off = 1; off < 16; off <<= 1)
            v = fminf(v, __shfl_xor(v, off, 32));
        if (l15 == 0) atomicMin(&rowmin[m0 + r + 8 * hi], fkey(v));
    }
}

// ---- stage 3: means of row/col mins, output max ----
__global__ __launch_bounds__(256) void finalize_kernel(
    const unsigned* __restrict__ rowmin,
    const unsigned* __restrict__ colmin,
    float* __restrict__ out)
{
    __shared__ float s1[256];
    __shared__ float s2[256];
    float a = 0.0f, b = 0.0f;
    for (int i = threadIdx.x; i < ROWS; i += 256) {
        a += fkey_inv(rowmin[i]);
        b += fkey_inv(colmin[i]);
    }
    s1[threadIdx.x] = a;
    s2[threadIdx.x] = b;
    __syncthreads();
    for (int off = 128; off > 0; off >>= 1) {
        if (threadIdx.x < off) {
            s1[threadIdx.x] += s1[threadIdx.x + off];
            s2[threadIdx.x] += s2[threadIdx.x + off];
        }
        __syncthreads();
    }
    if (threadIdx.x == 0)
        out[0] = fmaxf(s1[0] / (float)ROWS, s2[0] / (float)ROWS);
}

extern "C" void kernel_launch(void* const* d_in, const int* in_sizes, int n_in,
                              void* d_out, int out_size, void* d_ws, size_t ws_size,
                              hipStream_t stream) {
    const float* input_img  = (const float*)d_in[0];
    const float* target_img = (const float*)d_in[1];
    const int*   iy_in      = (const int*)d_in[2];
    const int*   ix_in      = (const int*)d_in[3];
    const int*   iy_tg      = (const int*)d_in[4];
    const int*   ix_tg      = (const int*)d_in[5];
    float* out = (float*)d_out;

    // Workspace layout (~6.5 MB total)
    float*    Xm     = (float*)d_ws;                       // ROWS*64
    float*    Ym     = Xm + (size_t)ROWS * KDIM;           // ROWS*64
    float*    xsq    = Ym + (size_t)ROWS * KDIM;           // ROWS
    float*    ysq    = xsq + ROWS;                         // ROWS
    float*    xinv   = ysq + ROWS;                         // ROWS
    float*    yinv   = xinv + ROWS;                        // ROWS
    unsigned* rowmin = (unsigned*)(yinv + ROWS);           // ROWS
    unsigned* colmin = rowmin + ROWS;                      // ROWS

    // Stage 0: init mins (rowmin & colmin are contiguous)
    init_min_kernel<<<(2 * ROWS + 255) / 256, 256, 0, stream>>>(rowmin, 2 * ROWS);

    // Stage 1: gather + YUV projection + row norms (1536 blocks * 8 waves = 12288 rows)
    gather_yuv_kernel<<<ROWS / 8, 256, 0, stream>>>(input_img,  iy_in, ix_in, Xm, xsq, xinv);
    gather_yuv_kernel<<<ROWS / 8, 256, 0, stream>>>(target_img, iy_tg, ix_tg, Ym, ysq, yinv);

    // Stage 2: fused WMMA distance + min reductions.
    // Waves = NT * (NT/NSUB) = 768 * 192 = 147456; 8 waves per block.
    gemm_dist_kernel<<<(NT * (NT / NSUB)) / 8, 256, 0, stream>>>(
        Xm, Ym, xsq, ysq, xinv, yinv, rowmin, colmin);

    // Stage 3: scalar result
    finalize_kernel<<<1, 256, 0, stream>>>(rowmin, colmin, out);
}